// Embedding_9423158247955
// MI455X (gfx1250) — compile-verified
//
#include <hip/hip_runtime.h>
#include <stdint.h>

// Problem shape (fixed by the reference)
constexpr int VOCAB = 50257;
constexpr int SEQ   = 2048;
constexpr int DIM   = 768;
constexpr int BATCH = 8;

typedef float float4v __attribute__((ext_vector_type(4)));

// One block per sequence position s.
// 192 threads; thread t owns d = 4*t .. 4*t+3 (float4 lane), for all 8 batches.
//
// out[b,s,d] = W_emb[d, tok[b,s]] + W_pos[s,d]
//
// Memory plan (MI455X):
//  * W_emb (154 MB) < 192 MB L2: gather is scattered 4B reads, keep it RT so it
//    stays L2-resident across graph replays.
//  * Output (50 MB) is a pure stream: non-temporal b128 stores so it does not
//    evict W_emb from L2.
//  * W_pos row (3 KB) pulled via the CDNA5 async engine
//    (global_load_async_to_lds_b128, ASYNCcnt) overlapped with the gathers;
//    reused 8x across the batch dimension.
//  * tokens[b*SEQ + s] are block-uniform -> scalar loads.
__global__ __launch_bounds__(DIM / 4) void embed_pos_kernel(
    const int*   __restrict__ tokens,  // [BATCH, SEQ]
    const float* __restrict__ W_emb,   // [DIM, VOCAB]
    const float* __restrict__ W_pos,   // [SEQ, DIM]
    float*       __restrict__ out)     // [BATCH, SEQ, DIM]
{
    __shared__ float posLds[DIM];

    const int s  = blockIdx.x;        // 0..SEQ-1
    const int t  = threadIdx.x;       // 0..191
    const int d0 = t * 4;

    // ---- Kick off async copy of this block's W_pos row into LDS ------------
    // 192 lanes x 16B = 3072B = one full row. Each lane later reads back only
    // its own 16 bytes, so no workgroup barrier is required, just the
    // ASYNCcnt wait. Issued first so its latency hides under the gathers.
    {
        const float* gsrc = W_pos + (size_t)s * DIM + d0;
        // Workgroup-relative LDS byte offset of this lane's 16B slot.
        uint32_t lds_off = (uint32_t)(uintptr_t)(&posLds[0]) + (uint32_t)t * 16u;
        asm volatile("global_load_async_to_lds_b128 %0, %1, off"
                     :
                     : "v"(lds_off), "v"(gsrc)
                     : "memory");
    }

    // ---- Block-uniform token ids (compiler scalarizes to s_load) -----------
    int tok[BATCH];
#pragma unroll
    for (int b = 0; b < BATCH; ++b)
        tok[b] = tokens[b * SEQ + s];

    // ---- Scattered gather: 4 d-rows x 8 tokens = 32 independent loads ------
    // Addresses W_emb[d*VOCAB + tok]: one 4B element per cacheline, served
    // from the 192MB L2 once W_emb is resident.
    const float* __restrict__ row0 = W_emb + (size_t)d0 * VOCAB;
    float4v emb[BATCH];
#pragma unroll
    for (int b = 0; b < BATCH; ++b) {
        const int tk = tok[b];
        emb[b].x = row0[tk];
        emb[b].y = row0[(size_t)VOCAB     + tk];
        emb[b].z = row0[(size_t)VOCAB * 2 + tk];
        emb[b].w = row0[(size_t)VOCAB * 3 + tk];
    }

    // ---- Wait for the async W_pos copy, read back own 16B from LDS ---------
    asm volatile("s_wait_asynccnt 0" ::: "memory");
    float4v pos = *(const float4v*)(&posLds[d0]);

    // ---- Add + streaming (non-temporal) stores: keep W_emb hot in L2 -------
#pragma unroll
    for (int b = 0; b < BATCH; ++b) {
        float4v v = emb[b] + pos;
        float4v* dst = (float4v*)(out + ((size_t)b * SEQ + s) * DIM + d0);
        __builtin_nontemporal_store(v, dst);
    }
}

extern "C" void kernel_launch(void* const* d_in, const int* in_sizes, int n_in,
                              void* d_out, int out_size, void* d_ws, size_t ws_size,
                              hipStream_t stream) {
    (void)in_sizes; (void)n_in; (void)out_size; (void)d_ws; (void)ws_size;
    const int*   tokens = (const int*)d_in[0];
    const float* W_emb  = (const float*)d_in[1];
    const float* W_pos  = (const float*)d_in[2];
    float*       out    = (float*)d_out;

    dim3 grid(SEQ);          // 2048 blocks, one per sequence position
    dim3 block(DIM / 4);     // 192 threads = 6 wave32
    embed_pos_kernel<<<grid, block, 0, stream>>>(tokens, W_emb, W_pos, out);
}